// STGNNLayer_9010841387389
// MI455X (gfx1250) — compile-verified
//
#include <hip/hip_runtime.h>
#include <hip/hip_bf16.h>

typedef float v2f __attribute__((ext_vector_type(2)));
typedef float v8f __attribute__((ext_vector_type(8)));
typedef int   v4i __attribute__((ext_vector_type(4)));

#define BSZ   4
#define TLEN  32
#define NN    1024
#define FF    128
#define KORD  4

#define MT   64      // node rows per block
#define MC   64      // K-chunk staged in LDS
#define LDA  68      // 68*4=272B rows: 16B aligned, banks skew 4/row
#define LDB  132     // 132*4=528B rows: 16B aligned, banks skew 4/row
#define LDT  132

#if defined(__gfx1250__) && __has_builtin(__builtin_amdgcn_global_load_async_to_lds_b128)
#define HAVE_ASYNC_LDS 1
#else
#define HAVE_ASYNC_LDS 0
#endif

// copy 16 bytes global -> LDS (async DMA path on CDNA5, tracked by ASYNCcnt)
static __device__ __forceinline__ void async_cp16(const float* g, float* l) {
#if HAVE_ASYNC_LDS
  typedef __attribute__((address_space(1))) v4i* gp_t;   // prints as int4 __device__*
  typedef __attribute__((address_space(3))) v4i* lp_t;   // LDS
  __builtin_amdgcn_global_load_async_to_lds_b128((gp_t)g, (lp_t)l, 0, 0);
#else
  const float4 v = *(const float4*)g;
  l[0] = v.x; l[1] = v.y; l[2] = v.z; l[3] = v.w;
#endif
}

static __device__ __forceinline__ void wait_async() {
#if HAVE_ASYNC_LDS
#if __has_builtin(__builtin_amdgcn_s_wait_asynccnt)
  __builtin_amdgcn_s_wait_asynccnt(0);
#else
  asm volatile("s_wait_asynccnt 0x0" ::: "memory");
#endif
#endif
}

static __device__ __forceinline__ v8f wmma_f32(v2f a, v2f b, v8f c) {
  // V_WMMA_F32_16X16X4_F32 : D = A(16x4) * B(4x16) + C, fp32, wave32
  return __builtin_amdgcn_wmma_f32_16x16x4_f32(false, a, false, b, (short)0, c,
                                               false, false);
}

// copy a 64x64 f32 tile (row stride src_ld) into LDS with stride LDA
static __device__ __forceinline__ void stage64x64(float* dst, const float* src,
                                                  int src_ld, int tid) {
  for (int jv = tid; jv < 64 * 16; jv += 256) {
    const int r  = jv >> 4;
    const int m4 = (jv & 15) << 2;
    async_cp16(src + (size_t)r * src_ld + m4, dst + r * LDA + m4);
  }
}

// copy a 64x128 f32 tile (row stride src_ld) into LDS with stride dld
static __device__ __forceinline__ void stage64x128(float* dst, int dld,
                                                   const float* src, int src_ld,
                                                   int tid) {
  for (int jv = tid; jv < 64 * 32; jv += 256) {
    const int r  = jv >> 5;
    const int f4 = (jv & 31) << 2;
    async_cp16(src + (size_t)r * src_ld + f4, dst + r * dld + f4);
  }
}

// C[1024x1024] = A @ B  (row-major, used for S^2 and S^3)
__global__ __launch_bounds__(256) void gemm1024_kernel(
    const float* __restrict__ A, const float* __restrict__ Bm,
    float* __restrict__ C) {
  __shared__ float smem[MT * LDA + MC * LDB];
  float* ldsA = smem;
  float* ldsB = smem + MT * LDA;

  const int tid  = threadIdx.x;
  const int wave = tid >> 5;
  const int lane = tid & 31;
  const int half = lane >> 4;
  const int l16  = lane & 15;
  const int row0 = blockIdx.x * MT;
  const int colB = blockIdx.y * 128;

  const v8f z8 = {0.f, 0.f, 0.f, 0.f, 0.f, 0.f, 0.f, 0.f};
  v8f acc[4] = {z8, z8, z8, z8};

  for (int u0 = 0; u0 < NN; u0 += MC) {
    stage64x64(ldsA, A + (size_t)row0 * NN + u0, NN, tid);
    stage64x128(ldsB, LDB, Bm + (size_t)u0 * NN + colB, NN, tid);
    wait_async();
    __syncthreads();
    for (int u = 0; u < MC; u += 4) {
      v2f bb;
      bb.x = ldsB[(u + 2 * half    ) * LDB + wave * 16 + l16];
      bb.y = ldsB[(u + 2 * half + 1) * LDB + wave * 16 + l16];
#pragma unroll
      for (int rt = 0; rt < 4; ++rt) {
        v2f aa;
        aa.x = ldsA[(rt * 16 + l16) * LDA + u + 2 * half];
        aa.y = ldsA[(rt * 16 + l16) * LDA + u + 2 * half + 1];
        acc[rt] = wmma_f32(aa, bb, acc[rt]);
      }
    }
    __syncthreads();
  }
#pragma unroll
  for (int rt = 0; rt < 4; ++rt)
#pragma unroll
    for (int i = 0; i < 8; ++i)
      C[(size_t)(row0 + rt * 16 + i + 8 * half) * NN + colB + wave * 16 + l16] =
          acc[rt][i];
}

// Wt[k][i][o] = W[o][i][k]  (so second-GEMM B loads are contiguous in o)
__global__ __launch_bounds__(256) void wtrans_kernel(const float* __restrict__ W,
                                                     float* __restrict__ Wt) {
  const int tid = blockIdx.x * 256 + threadIdx.x;  // 16384 = FF*FF threads
  const int i = tid >> 7;
  const int o = tid & 127;
  const float4 w4 = *(const float4*)(W + ((size_t)o * FF + i) * KORD);
  Wt[0 * FF * FF + i * FF + o] = w4.x;
  Wt[1 * FF * FF + i * FF + o] = w4.y;
  Wt[2 * FF * FF + i * FF + o] = w4.z;
  Wt[3 * FF * FF + i * FF + o] = w4.w;
}

// Fused: Y[b,t,rows64,:] = bias + sum_k (P_k[rows64]@X[b,t-k]) @ Wk^T
__global__ __launch_bounds__(256) void stgnn_kernel(
    const float* __restrict__ X, const float* __restrict__ S,
    const float* __restrict__ S2, const float* __restrict__ S3,
    const float* __restrict__ Wt, const float* __restrict__ bias,
    float* __restrict__ Y) {
  __shared__ float smem[MT * LDA + MC * LDB];
  float* ldsA = smem;
  float* ldsB = smem + MT * LDA;
  float* ldsT = smem;  // aliases A/B region (64*132 <= 64*68+64*132); barriered

  const int tid  = threadIdx.x;
  const int wave = tid >> 5;
  const int lane = tid & 31;
  const int half = lane >> 4;
  const int l16  = lane & 15;
  const int bt = blockIdx.x;            // 0..127
  const int b = bt / TLEN, t = bt % TLEN;
  const int row0 = blockIdx.y * MT;

  const v8f z8 = {0.f, 0.f, 0.f, 0.f, 0.f, 0.f, 0.f, 0.f};
  v8f acc[4] = {z8, z8, z8, z8};

  for (int k = 0; k < KORD; ++k) {
    const int ts = t - k;
    if (ts < 0) break;  // uniform across block
    const float* Xs = X + (size_t)(b * TLEN + ts) * NN * FF;
    __syncthreads();  // previous k's readers of ldsT are done before restaging
    if (k == 0) {
      // T_0 = X rows directly
      stage64x128(ldsT, LDT, Xs + (size_t)row0 * FF, FF, tid);
      wait_async();
      __syncthreads();
    } else {
      const float* P = (k == 1) ? S : (k == 2) ? S2 : S3;
      v8f acc2[4] = {z8, z8, z8, z8};
      for (int u0 = 0; u0 < NN; u0 += MC) {
        stage64x64(ldsA, P + (size_t)row0 * NN + u0, NN, tid);
        stage64x128(ldsB, LDB, Xs + (size_t)u0 * FF, FF, tid);
        if (u0 + MC < NN)  // warm L2/WGP$ for the next X slab
          __builtin_prefetch(Xs + (size_t)(u0 + MC) * FF + tid * 16, 0, 1);
        wait_async();
        __syncthreads();
        for (int u = 0; u < MC; u += 4) {
          v2f bb;
          bb.x = ldsB[(u + 2 * half    ) * LDB + wave * 16 + l16];
          bb.y = ldsB[(u + 2 * half + 1) * LDB + wave * 16 + l16];
#pragma unroll
          for (int rt = 0; rt < 4; ++rt) {
            v2f aa;
            aa.x = ldsA[(rt * 16 + l16) * LDA + u + 2 * half];
            aa.y = ldsA[(rt * 16 + l16) * LDA + u + 2 * half + 1];
            acc2[rt] = wmma_f32(aa, bb, acc2[rt]);
          }
        }
        __syncthreads();
      }
      // park T_k = P_k @ X in LDS for the W-apply GEMM
#pragma unroll
      for (int rt = 0; rt < 4; ++rt)
#pragma unroll
        for (int i = 0; i < 8; ++i)
          ldsT[(rt * 16 + i + 8 * half) * LDT + wave * 16 + l16] = acc2[rt][i];
      __syncthreads();
    }
    // acc += T_k(64x128) @ Wk^T(128x128), B read from pre-transposed Wt[k][i][o]
    const float* Wk = Wt + (size_t)k * FF * FF;
    for (int ii = 0; ii < FF; ii += 4) {
      v2f bb;
      bb.x = Wk[(ii + 2 * half    ) * FF + wave * 16 + l16];
      bb.y = Wk[(ii + 2 * half + 1) * FF + wave * 16 + l16];
#pragma unroll
      for (int rt = 0; rt < 4; ++rt) {
        v2f aa;
        aa.x = ldsT[(rt * 16 + l16) * LDT + ii + 2 * half];
        aa.y = ldsT[(rt * 16 + l16) * LDT + ii + 2 * half + 1];
        acc[rt] = wmma_f32(aa, bb, acc[rt]);
      }
    }
  }

  const float bv = bias[wave * 16 + l16];
  float* Yp = Y + (size_t)(b * TLEN + t) * NN * FF;
#pragma unroll
  for (int rt = 0; rt < 4; ++rt)
#pragma unroll
    for (int i = 0; i < 8; ++i)
      Yp[(size_t)(row0 + rt * 16 + i + 8 * half) * FF + wave * 16 + l16] =
          acc[rt][i] + bv;
}

extern "C" void kernel_launch(void* const* d_in, const int* in_sizes, int n_in,
                              void* d_out, int out_size, void* d_ws,
                              size_t ws_size, hipStream_t stream) {
  (void)in_sizes; (void)n_in; (void)out_size; (void)ws_size;
  const float* X    = (const float*)d_in[0];
  const float* S    = (const float*)d_in[1];
  const float* W    = (const float*)d_in[2];
  const float* bias = (const float*)d_in[3];
  float* Y = (float*)d_out;

  // workspace layout: S2 (4MB) | S3 (4MB) | Wt (256KB)  -> 8.25 MB total
  float* S2 = (float*)d_ws;
  float* S3 = S2 + (size_t)NN * NN;
  float* Wt = S3 + (size_t)NN * NN;

  dim3 gg(NN / MT, NN / 128);
  gemm1024_kernel<<<gg, 256, 0, stream>>>(S, S, S2);    // S^2
  gemm1024_kernel<<<gg, 256, 0, stream>>>(S2, S, S3);   // S^3
  wtrans_kernel<<<(FF * FF) / 256, 256, 0, stream>>>(W, Wt);

  dim3 gm(BSZ * TLEN, NN / MT);
  stgnn_kernel<<<gm, 256, 0, stream>>>(X, S, S2, S3, Wt, bias, Y);
}